// LearnablePatchErasing_60662118088856
// MI455X (gfx1250) — compile-verified
//
#include <hip/hip_runtime.h>

// LearnablePatchErasing: out = clip((1-a^2)*img + a^2*noise, -1, 1)
// img  [256,3,224,224] fp32, noise [3,224,224] fp32, alpha [196] fp32
// Pure streaming kernel: HBM-bound (308 MB -> ~13.2 us floor @ 23.3 TB/s).
// gfx1250 paths used: global_load_async_to_lds_b128 (ASYNCcnt), s_wait_asynccnt,
// ds_load_b128 (LDS-resident noise row), non-temporal b128 load/store,
// global_prefetch_b8.

#define NB 256
#define NC 3
#define HW 224
#define PLANE (HW * HW)        // 50176
#define IMG_SZ (NC * PLANE)    // 150528
#define HP 14                  // patches per side

typedef float v4f __attribute__((ext_vector_type(4)));

__global__ __launch_bounds__(224) void patch_erase_kernel(
    const float* __restrict__ img,
    const float* __restrict__ noise,
    const float* __restrict__ alpha,
    float* __restrict__ out)
{
    __shared__ float s_noise[HW];   // one noise row, shared by all 256 batches

    const int bid  = blockIdx.x;            // [0, 672) -> (c, h)
    const int c    = bid / HW;
    const int h    = bid - c * HW;
    const int tid  = threadIdx.x;           // [0, 224) = 7 waves (wave32)
    const int slot = tid % 56;              // float4 slot within the row
    const int bg   = tid / 56;              // batch group [0, 4)
    const int w    = slot * 4;              // 16B-aligned, never straddles a 16px patch

    const long long rowbase = (long long)c * PLANE + (long long)h * HW;

    // ---- Stage the shared noise row into LDS via CDNA5 async copy ----------
    if (bg == 0) {
        unsigned lds_off = (unsigned)(unsigned long long)(&s_noise[w]);   // low 32 bits = LDS byte offset
        unsigned long long gaddr = (unsigned long long)(noise + rowbase + w);
        asm volatile("global_load_async_to_lds_b128 %0, %1, off"
                     :: "v"(lds_off), "v"(gaddr) : "memory");
    }
    asm volatile("s_wait_asynccnt 0" ::: "memory");   // issuing waves drain ASYNCcnt
    __syncthreads();                                   // publish LDS to all waves

    // ---- Hoist per-thread constants out of the batch loop ------------------
    const float a   = alpha[(h >> 4) * HP + (w >> 4)];
    const float a2  = a * a;
    const float inv = 1.0f - a2;

    v4f nz = *(const v4f*)(&s_noise[w]);   // ds_load_b128
    v4f pn;
    pn.x = a2 * nz.x; pn.y = a2 * nz.y; pn.z = a2 * nz.z; pn.w = a2 * nz.w;

    const float* gin  = img + (size_t)bg * IMG_SZ + rowbase + w;
    float*       gout = out + (size_t)bg * IMG_SZ + rowbase + w;
    const size_t stride = (size_t)4 * IMG_SZ;   // 4 batches per step

    // ---- Streaming loop: 64 iterations of NT b128 load / fma / clamp / NT b128 store
    #pragma unroll 4
    for (int b = bg; b < NB; b += 4) {
        __builtin_prefetch(gin + 2 * stride, 0, 0);          // global_prefetch_b8, 2 steps ahead
        v4f v = __builtin_nontemporal_load((const v4f*)gin); // TH=NT: img streamed once
        v4f r;
        r.x = fminf(fmaxf(__builtin_fmaf(inv, v.x, pn.x), -1.0f), 1.0f);
        r.y = fminf(fmaxf(__builtin_fmaf(inv, v.y, pn.y), -1.0f), 1.0f);
        r.z = fminf(fmaxf(__builtin_fmaf(inv, v.z, pn.z), -1.0f), 1.0f);
        r.w = fminf(fmaxf(__builtin_fmaf(inv, v.w, pn.w), -1.0f), 1.0f);
        __builtin_nontemporal_store(r, (v4f*)gout);          // TH=NT: out never re-read
        gin  += stride;
        gout += stride;
    }
}

extern "C" void kernel_launch(void* const* d_in, const int* in_sizes, int n_in,
                              void* d_out, int out_size, void* d_ws, size_t ws_size,
                              hipStream_t stream) {
    const float* img   = (const float*)d_in[0];
    const float* noise = (const float*)d_in[1];
    const float* alpha = (const float*)d_in[2];
    float*       out   = (float*)d_out;

    dim3 grid(NC * HW);   // 672 blocks: one per (channel, row)
    dim3 block(224);      // 56 float4 slots x 4 batch groups = 7 wave32s

    hipLaunchKernelGGL(patch_erase_kernel, grid, block, 0, stream,
                       img, noise, alpha, out);
}